// MTGHead_35802847380251
// MI455X (gfx1250) — compile-verified
//
#include <hip/hip_runtime.h>

// ---------------------------------------------------------------------------
// MTG attention head for MI455X (gfx1250), fp32 via V_WMMA_F32_16X16X4_F32.
//   x  [8,2048,1024]  Wq/Wk/Wv [1024,64]
//   out[8,2048,64], wei[8,2048,2048]   (d_out = out ++ wei, fp32)
// Projection pass uses the Tensor Data Mover (tensor_load_to_lds) to stage
// x tiles; attention pass holds the full 32x2048 score strip in LDS (256 KB
// of the 320 KB/WGP CDNA5 LDS) so wei is streamed to HBM exactly once with
// non-temporal stores.
// ---------------------------------------------------------------------------

typedef __attribute__((ext_vector_type(2))) float v2f;
typedef __attribute__((ext_vector_type(4))) float v4f;
typedef __attribute__((ext_vector_type(8))) float v8f;
typedef __attribute__((ext_vector_type(4))) unsigned int u32x4;
typedef __attribute__((ext_vector_type(8))) unsigned int u32x8;

#define B_   8
#define T_   2048
#define C_   1024
#define H_   64
#define M_   (B_ * T_)          // 16384 total rows
#define EPS_ 1e-5f
#define SCALE_ 0.125f           // 1/sqrt(64)

static __device__ __forceinline__ v8f wmma_f32(v2f a, v2f b, v8f c) {
  return __builtin_amdgcn_wmma_f32_16x16x4_f32(
      /*neg_a=*/false, a, /*neg_b=*/false, b,
      /*c_mod=*/(short)0, c, /*reuse_a=*/false, /*reuse_b=*/false);
}

// ---------------------------------------------------------------------------
// Kernel 1: QKV projection.  grid = (M/64, 3), block = 256 (8 waves).
// Each workgroup: 64 rows x 64 cols of one of {q,k,v}; K=1024 in LDS chunks.
// x tiles staged via TDM (tensor_load_to_lds); which==2 (v) stored transposed.
// ---------------------------------------------------------------------------
__global__ __launch_bounds__(256)
void mtg_qkv_proj(const float* __restrict__ x,
                  const float* __restrict__ Wq,
                  const float* __restrict__ Wk,
                  const float* __restrict__ Wv,
                  float* __restrict__ qo,   // [M, H]
                  float* __restrict__ ko,   // [M, H]
                  float* __restrict__ vT)   // [B][H][T]
{
  __shared__ float xs[64][64];    // x tile  (rows x K-chunk), TDM destination
  __shared__ float wsT[64][64];   // W chunk transposed: wsT[n][k]

  const int which   = blockIdx.y;            // 0=q 1=k 2=v
  const float* W    = (which == 0) ? Wq : (which == 1) ? Wk : Wv;
  const int rowBase = blockIdx.x * 64;
  const int tid  = threadIdx.x;
  const int lane = tid & 31;
  const int half = lane >> 4;                // 0: K+0/1, 1: K+2/3
  const int l16  = lane & 15;
  // wave id as a compiler-known scalar (TDM issue must be a scalar branch:
  // tensor ops ignore EXEC, so EXEC-predication would fire once per wave).
  const int swave = __builtin_amdgcn_readfirstlane(tid >> 5);

  // two 16x16 output tiles per wave (4x4 tile grid)
  const int t0 = swave * 2, t1 = swave * 2 + 1;
  const int tr0 = t0 >> 2, tc0 = t0 & 3;
  const int tr1 = t1 >> 2, tc1 = t1 & 3;
  v8f acc0 = {}; v8f acc1 = {};

  // ---- invariant D# fields (wave-uniform -> SGPRs) ------------------------
  const unsigned lds_xs = (unsigned)(unsigned long long)(uintptr_t)&xs[0][0];
  u32x8 g1;
  g1[0] = (2u << 16);                    // workgroup_mask=0, data_size=4B
  g1[1] = ((unsigned)C_ & 0xFFFFu) << 16;          // tensor_dim0[15:0]
  g1[2] = ((unsigned)C_ >> 16) | (((unsigned)M_ & 0xFFFFu) << 16); // dim0 hi | dim1 lo
  g1[3] = ((unsigned)M_ >> 16) | (64u << 16);      // dim1 hi | tile_dim0=64
  g1[4] = 64u;                                     // tile_dim1=64, tile_dim2=0
  g1[5] = (unsigned)C_;                            // tensor_dim0_stride lo
  g1[6] = 0u;                                      // stride0 hi | stride1 lo
  g1[7] = 0u;                                      // stride1 hi

  for (int kk = 0; kk < C_; kk += 64) {
    // ---- TDM: stage x[rowBase..+64)[kk..+64) -> xs, one issue per block ---
    if (swave == 0) {
      unsigned long long gaddr =
          (unsigned long long)(uintptr_t)x +
          ((unsigned long long)rowBase * C_ + (unsigned)kk) * 4ull;
      u32x4 g0;
      g0[0] = 1u;                                  // count=1, normal mode
      g0[1] = lds_xs;                              // LDS byte address
      g0[2] = (unsigned)(gaddr & 0xFFFFFFFFull);   // global_addr[31:0]
      g0[3] = (unsigned)((gaddr >> 32) & 0x1FFFFFFull) | (2u << 30); // type=2
      asm volatile("tensor_load_to_lds %0, %1" :: "s"(g0), "s"(g1) : "memory");
      __builtin_amdgcn_s_wait_tensorcnt(0);
    }
    // stage W chunk transposed (so B-fragments are contiguous b64 reads)
    for (int i = tid; i < 64 * 64; i += 256) {
      int kr = i >> 6, n = i & 63;
      wsT[n][kr] = W[(size_t)(kk + kr) * H_ + n];
    }
    __syncthreads();

    for (int k4 = 0; k4 < 64; k4 += 4) {
      const int kc = k4 + 2 * half;
      v2f a0 = *(const v2f*)&xs[tr0 * 16 + l16][kc];
      v2f b0 = *(const v2f*)&wsT[tc0 * 16 + l16][kc];
      acc0 = wmma_f32(a0, b0, acc0);
      v2f a1 = *(const v2f*)&xs[tr1 * 16 + l16][kc];
      v2f b1 = *(const v2f*)&wsT[tc1 * 16 + l16][kc];
      acc1 = wmma_f32(a1, b1, acc1);
    }
    __syncthreads();   // fences xs readers before next TDM overwrite
  }

  // store C/D tiles: element (m,n) lives in VGPR (m%8), lane (m/8)*16 + n
#pragma unroll
  for (int r = 0; r < 8; ++r) {
    int row0 = tr0 * 16 + half * 8 + r, col0 = tc0 * 16 + l16;
    int row1 = tr1 * 16 + half * 8 + r, col1 = tc1 * 16 + l16;
    int g0r = rowBase + row0, g1r = rowBase + row1;
    if (which == 0) {
      qo[(size_t)g0r * H_ + col0] = acc0[r];
      qo[(size_t)g1r * H_ + col1] = acc1[r];
    } else if (which == 1) {
      ko[(size_t)g0r * H_ + col0] = acc0[r];
      ko[(size_t)g1r * H_ + col1] = acc1[r];
    } else {
      int b0b = g0r / T_, t0t = g0r % T_;
      int b1b = g1r / T_, t1t = g1r % T_;
      vT[((size_t)b0b * H_ + col0) * T_ + t0t] = acc0[r];
      vT[((size_t)b1b * H_ + col1) * T_ + t1t] = acc1[r];
    }
  }
}

// ---------------------------------------------------------------------------
// Kernel 2: fused attention.  grid = (T/32, B), block = 256 (8 waves).
// Dynamic LDS: sc[32][2048] (256 KB) + qs[32][64] (8 KB).
// ---------------------------------------------------------------------------
__global__ __launch_bounds__(256)
void mtg_attn(const float* __restrict__ q,    // [M,H]
              const float* __restrict__ k,    // [M,H]
              const float* __restrict__ vT,   // [B][H][T]
              float* __restrict__ outp,       // [B,T,H]
              float* __restrict__ wei)        // [B,T,T]
{
  extern __shared__ float smem[];
  float* sc = smem;                 // [32][T_]
  float* qs = smem + 32 * T_;       // [32][H_]
  __shared__ float redS[32][8], redQ[32][8], redM[32][8];
  __shared__ float rowMean[32], rowRstd[32], rowNmax[32], rowInv[32];

  const int b   = blockIdx.y;
  const int q0  = blockIdx.x * 32;
  const int tid = threadIdx.x;
  const int lane = tid & 31;
  const int wave = tid >> 5;
  const int half = lane >> 4;
  const int l16  = lane & 15;

  // stage q tile (32 x 64)
  for (int i = tid; i < 32 * (H_ / 4); i += 256) {
    int r = i >> 4, c4 = (i & 15) * 4;
    *(v4f*)&qs[r * H_ + c4] =
        *(const v4f*)&q[((size_t)b * T_ + q0 + r) * H_ + c4];
  }
  __syncthreads();

  // ---- Phase 1: raw scores S = (q_tile @ k^T) * scale  -> LDS -------------
  const float* kb = k + (size_t)b * T_ * H_;
  for (int j = wave; j < T_ / 16; j += 8) {     // 128 key-column tiles
    v8f c0 = {}; v8f c1 = {};
    const float* kp = kb + (size_t)(j * 16 + l16) * H_;  // key row for lane
    for (int k4 = 0; k4 < H_; k4 += 4) {
      const int kc = k4 + 2 * half;
      v2f bf = *(const v2f*)&kp[kc];                      // B[kc..kc+1][n]
      v2f a0 = *(const v2f*)&qs[l16 * H_ + kc];           // rows 0..15
      v2f a1 = *(const v2f*)&qs[(16 + l16) * H_ + kc];    // rows 16..31
      c0 = wmma_f32(a0, bf, c0);
      c1 = wmma_f32(a1, bf, c1);
    }
#pragma unroll
    for (int r = 0; r < 8; ++r) {
      int col = j * 16 + l16;
      sc[(half * 8 + r) * T_ + col]        = c0[r] * SCALE_;
      sc[(16 + half * 8 + r) * T_ + col]   = c1[r] * SCALE_;
    }
  }
  __syncthreads();

  // ---- Phase 1b: per-row sum / sumsq / max (8 threads per row) ------------
  {
    int r = tid >> 3, seg = tid & 7;
    const float* rowp = &sc[r * T_ + seg * 256];
    float s = 0.f, sq = 0.f, mx = -1e30f;
    for (int c = 0; c < 256; ++c) {
      float v = rowp[c];
      s += v; sq += v * v; mx = fmaxf(mx, v);
    }
    redS[r][seg] = s; redQ[r][seg] = sq; redM[r][seg] = mx;
  }
  __syncthreads();
  if ((tid & 7) == 0) {
    int r = tid >> 3;
    float s = 0.f, sq = 0.f, mx = -1e30f;
    for (int i = 0; i < 8; ++i) {
      s += redS[r][i]; sq += redQ[r][i]; mx = fmaxf(mx, redM[r][i]);
    }
    float mean = s * (1.0f / T_);
    float var  = sq * (1.0f / T_) - mean * mean;   // == mean of centered^2
    float rstd = rsqrtf(var + EPS_);
    rowMean[r] = mean;
    rowRstd[r] = rstd;
    rowNmax[r] = (mx - mean) * rstd;               // max of normalized row
  }
  __syncthreads();

  // ---- Phase 2: exp((s-mean)*rstd - nmax) in place + row denominators ----
  {
    int r = tid >> 3, seg = tid & 7;
    float mean = rowMean[r], rstd = rowRstd[r], nmax = rowNmax[r];
    float* rowp = &sc[r * T_ + seg * 256];
    float s = 0.f;
    for (int c = 0; c < 256; ++c) {
      float e = __expf((rowp[c] - mean) * rstd - nmax);
      rowp[c] = e;
      s += e;
    }
    redS[r][seg] = s;
  }
  __syncthreads();
  if ((tid & 7) == 0) {
    int r = tid >> 3;
    float s = 0.f;
    for (int i = 0; i < 8; ++i) s += redS[r][i];
    rowInv[r] = 1.0f / s;
  }
  __syncthreads();

  // ---- Phase 3: out = softmax(S) @ V  (P unnormalized in LDS, scale last) -
  {
    const int m = wave >> 2, n = wave & 3;        // 2x4 tile grid, 1 per wave
    const float* vTp = vT + (size_t)b * H_ * T_;
    const float* vcol = vTp + (size_t)(n * 16 + l16) * T_;  // v^T row = col n
    __builtin_prefetch(vcol, 0, 0);
    v8f acc = {};
    for (int k4 = 0; k4 < T_; k4 += 4) {
      const int kc = k4 + 2 * half;
      v2f a = *(const v2f*)&sc[(m * 16 + l16) * T_ + kc];   // P fragment
      v2f bf = *(const v2f*)&vcol[kc];                      // V fragment
      acc = wmma_f32(a, bf, acc);
    }
#pragma unroll
    for (int r = 0; r < 8; ++r) {
      int row = m * 16 + half * 8 + r;
      int col = n * 16 + l16;
      outp[((size_t)b * T_ + q0 + row) * H_ + col] = acc[r] * rowInv[row];
    }
  }

  // ---- Phase 4: stream wei = P * rowInv to HBM (non-temporal float4) ------
  for (int i = tid; i < 32 * (T_ / 4); i += 256) {
    int r = i >> 9;                 // T_/4 = 512 float4 per row
    int c4 = (i & 511) * 4;
    v4f p = *(const v4f*)&sc[r * T_ + c4];
    float inv = rowInv[r];
    p.x *= inv; p.y *= inv; p.z *= inv; p.w *= inv;
    __builtin_nontemporal_store(p, (v4f*)&wei[((size_t)b * T_ + q0 + r) * T_ + c4]);
  }
}

// ---------------------------------------------------------------------------
extern "C" void kernel_launch(void* const* d_in, const int* in_sizes, int n_in,
                              void* d_out, int out_size, void* d_ws, size_t ws_size,
                              hipStream_t stream) {
  (void)in_sizes; (void)n_in; (void)out_size; (void)ws_size;
  const float* x  = (const float*)d_in[0];
  const float* Wq = (const float*)d_in[1];
  const float* Wk = (const float*)d_in[2];
  const float* Wv = (const float*)d_in[3];

  float* qb = (float*)d_ws;              // [M,H]
  float* kb = qb + (size_t)M_ * H_;      // [M,H]
  float* vT = kb + (size_t)M_ * H_;      // [B][H][T]

  float* outp = (float*)d_out;                       // [B,T,H]
  float* wei  = outp + (size_t)M_ * H_;              // [B,T,T]

  hipLaunchKernelGGL(mtg_qkv_proj, dim3(M_ / 64, 3), dim3(256), 0, stream,
                     x, Wq, Wk, Wv, qb, kb, vT);

  const size_t lds_bytes = (size_t)(32 * T_ + 32 * H_) * sizeof(float); // 264 KB
  (void)hipFuncSetAttribute((const void*)mtg_attn,
                            hipFuncAttributeMaxDynamicSharedMemorySize,
                            (int)lds_bytes);
  hipLaunchKernelGGL(mtg_attn, dim3(T_ / 32, B_), dim3(256), lds_bytes, stream,
                     qb, kb, vT, outp, wei);
}